// FlattenAdditiveMul_33509334843953
// MI455X (gfx1250) — compile-verified
//
#include <hip/hip_runtime.h>
#include <math.h>

// CDNA5 / gfx1250. wave32. WMMA f32 16x16x4.
typedef __attribute__((ext_vector_type(2))) float v2f;
typedef __attribute__((ext_vector_type(4))) float v4f;
typedef __attribute__((ext_vector_type(8))) float v8f;

#define HEADS 8
#define DH 32
#define NNODES 50000

// ---------------------------------------------------------------------------
// Kernel 1: scores[e][h] = relu(q[e,h,:].a_q[h,:] + k[e,h,:].a_k[h,:])
// One wave handles 16 edges x 8 heads via V_WMMA_F32_16X16X4_F32.
// K-partition remap: chunk kc, lane-half kh covers d = kh*16 + kc*2 + {0,1},
// so each lane's A data for chunks (2j, 2j+1) is one contiguous b128 load.
// B is block-diagonal (only column h nonzero); every lane preloads the attn
// row of its own column (n&7) once, then per-head selects it in (no branches).
//   A-lane layout: lane l = 16*kh + m holds A[m][kh*2+v], v=0,1
//   B-lane layout: lane l = 16*kh + n holds B[kh*2+v][n]
//   D-lane layout: vgpr r, lane l -> D[r + 8*(l>>4)][l&15]
// ---------------------------------------------------------------------------
__global__ __launch_bounds__(256) void score_wmma_kernel(
    const float* __restrict__ q, const float* __restrict__ k,
    const float* __restrict__ attn, float* __restrict__ scores, int E)
{
  __shared__ __align__(16) float lds[8][256];  // 8 waves * 16x16 tile
  const int lane = threadIdx.x & 31;
  const int wave = threadIdx.x >> 5;
  const int e0   = (blockIdx.x * 8 + wave) * 16;
  if (e0 >= E) return;            // uniform per wave; grid exact for E=400000

  const int n  = lane & 15;       // A row (edge) / B,D column index
  const int kh = lane >> 4;       // K half

  const float* qrow = q + (size_t)(e0 + n) * (HEADS * DH);
  const float* krow = k + (size_t)(e0 + n) * (HEADS * DH);
  const float* arow = attn + (n & 7) * (2 * DH) + kh * 16;  // lanes n>=8: dup, always masked off

  // Preload this lane's B data: 16 floats a_q + 16 floats a_k (branch-free).
  v4f BQ[4], BK[4];
#pragma unroll
  for (int j = 0; j < 4; ++j) {
    BQ[j] = *(const v4f*)(arow + j * 4);
    BK[j] = *(const v4f*)(arow + DH + j * 4);
  }

  v8f accq = {0.f, 0.f, 0.f, 0.f, 0.f, 0.f, 0.f, 0.f};
  v8f acck = {0.f, 0.f, 0.f, 0.f, 0.f, 0.f, 0.f, 0.f};
  const v2f z2 = {0.f, 0.f};

#pragma unroll
  for (int h = 0; h < HEADS; ++h) {
    const bool mine = (n == h);          // this lane's column is head h
    const float* qh = qrow + h * DH + kh * 16;
    const float* kp = krow + h * DH + kh * 16;
#pragma unroll
    for (int j = 0; j < 4; ++j) {        // chunks kc = 2j, 2j+1
      v4f aq = *(const v4f*)(qh + j * 4);
      v4f ak = *(const v4f*)(kp + j * 4);
      v2f bq0 = mine ? __builtin_shufflevector(BQ[j], BQ[j], 0, 1) : z2;
      v2f bq1 = mine ? __builtin_shufflevector(BQ[j], BQ[j], 2, 3) : z2;
      v2f bk0 = mine ? __builtin_shufflevector(BK[j], BK[j], 0, 1) : z2;
      v2f bk1 = mine ? __builtin_shufflevector(BK[j], BK[j], 2, 3) : z2;
      accq = __builtin_amdgcn_wmma_f32_16x16x4_f32(
          false, __builtin_shufflevector(aq, aq, 0, 1), false, bq0, (short)0, accq, false, false);
      accq = __builtin_amdgcn_wmma_f32_16x16x4_f32(
          false, __builtin_shufflevector(aq, aq, 2, 3), false, bq1, (short)0, accq, false, false);
      acck = __builtin_amdgcn_wmma_f32_16x16x4_f32(
          false, __builtin_shufflevector(ak, ak, 0, 1), false, bk0, (short)0, acck, false, false);
      acck = __builtin_amdgcn_wmma_f32_16x16x4_f32(
          false, __builtin_shufflevector(ak, ak, 2, 3), false, bk1, (short)0, acck, false, false);
    }
  }

  // relu + transpose D through LDS so stores are contiguous float4s
  float* sl = &lds[wave][0];
#pragma unroll
  for (int r = 0; r < 8; ++r) {
    float v = accq[r] + acck[r];
    v = v > 0.f ? v : 0.f;
    sl[(r + 8 * kh) * 16 + n] = v;  // sl[edge*16 + head]
  }
  __syncthreads();

  const int er = lane >> 1;         // edge row 0..15
  const int hb = (lane & 1) * 4;    // head base 0 or 4
  float4 vv = *(const float4*)&sl[er * 16 + hb];
  *(float4*)(scores + (size_t)(e0 + er) * HEADS + hb) = vv;
}

// ---------------------------------------------------------------------------
// Kernel 2: per node, binary-search the sorted index for its edge range,
// compute per-head max and 1/sum(exp(s-max)).  Deterministic (no atomics).
// ---------------------------------------------------------------------------
__global__ __launch_bounds__(256) void seg_reduce_kernel(
    const int* __restrict__ index, const float* __restrict__ scores,
    float* __restrict__ seg_max, float* __restrict__ seg_inv,
    int E, int nnodes)
{
  int node = blockIdx.x * blockDim.x + threadIdx.x;
  if (node >= nnodes) return;

  int lo = 0, hi = E;
  while (lo < hi) { int mid = (lo + hi) >> 1; if (index[mid] < node) lo = mid + 1; else hi = mid; }
  const int start = lo;
  hi = E;
  while (lo < hi) { int mid = (lo + hi) >> 1; if (index[mid] < node + 1) lo = mid + 1; else hi = mid; }
  const int end = lo;

  float mx[HEADS], sm[HEADS];
#pragma unroll
  for (int h = 0; h < HEADS; ++h) { mx[h] = -__builtin_inff(); sm[h] = 0.f; }

  for (int e = start; e < end; ++e) {
    const float4* row = (const float4*)(scores + (size_t)e * HEADS);
    float4 r0 = row[0], r1 = row[1];
    mx[0] = fmaxf(mx[0], r0.x); mx[1] = fmaxf(mx[1], r0.y);
    mx[2] = fmaxf(mx[2], r0.z); mx[3] = fmaxf(mx[3], r0.w);
    mx[4] = fmaxf(mx[4], r1.x); mx[5] = fmaxf(mx[5], r1.y);
    mx[6] = fmaxf(mx[6], r1.z); mx[7] = fmaxf(mx[7], r1.w);
  }
  for (int e = start; e < end; ++e) {
    const float4* row = (const float4*)(scores + (size_t)e * HEADS);
    float4 r0 = row[0], r1 = row[1];
    sm[0] += expf(r0.x - mx[0]); sm[1] += expf(r0.y - mx[1]);
    sm[2] += expf(r0.z - mx[2]); sm[3] += expf(r0.w - mx[3]);
    sm[4] += expf(r1.x - mx[4]); sm[5] += expf(r1.y - mx[5]);
    sm[6] += expf(r1.z - mx[6]); sm[7] += expf(r1.w - mx[7]);
  }
#pragma unroll
  for (int h = 0; h < HEADS; ++h) {
    seg_max[(size_t)node * HEADS + h] = mx[h];
    seg_inv[(size_t)node * HEADS + h] = (sm[h] > 0.f) ? 1.f / sm[h] : 0.f;
  }
}

// ---------------------------------------------------------------------------
// Kernel 3: out[e,h] = exp(score - seg_max[index[e],h]) * seg_inv[index[e],h]
// (scores live in d_out; overwrite in place, element-wise)
// ---------------------------------------------------------------------------
__global__ __launch_bounds__(256) void finalize_kernel(
    const int* __restrict__ index, const float* __restrict__ seg_max,
    const float* __restrict__ seg_inv, float* __restrict__ out, int total)
{
  int i = blockIdx.x * blockDim.x + threadIdx.x;
  if (i >= total) return;
  int e = i >> 3, h = i & 7;
  int node = index[e];
  float s = out[i];
  out[i] = expf(s - seg_max[(size_t)node * HEADS + h]) *
           seg_inv[(size_t)node * HEADS + h];
}

extern "C" void kernel_launch(void* const* d_in, const int* in_sizes, int n_in,
                              void* d_out, int out_size, void* d_ws, size_t ws_size,
                              hipStream_t stream) {
  (void)n_in; (void)out_size; (void)ws_size;
  const float* q     = (const float*)d_in[0];
  const float* k     = (const float*)d_in[1];
  const float* attn  = (const float*)d_in[2];
  const int*   index = (const int*)d_in[3];
  const int E = in_sizes[3];            // 400000
  const int nnodes = NNODES;            // 50000 (fixed by reference)

  float* scores  = (float*)d_out;                       // scores then output, in place
  float* seg_max = (float*)d_ws;                        // nnodes*8 floats
  float* seg_inv = seg_max + (size_t)nnodes * HEADS;    // nnodes*8 floats

  const int tiles = (E + 15) / 16;
  score_wmma_kernel<<<dim3((tiles + 7) / 8), dim3(256), 0, stream>>>(
      q, k, attn, scores, E);

  seg_reduce_kernel<<<dim3((nnodes + 255) / 256), dim3(256), 0, stream>>>(
      index, scores, seg_max, seg_inv, E, nnodes);

  const int total = E * HEADS;
  finalize_kernel<<<dim3((total + 255) / 256), dim3(256), 0, stream>>>(
      index, seg_max, seg_inv, scores, total);
}